// DGAD_net_76124000354384
// MI455X (gfx1250) — compile-verified
//
#include <hip/hip_runtime.h>

// ---------------------------------------------------------------------------
// DGAD net forward on MI455X (gfx1250, wave32).
// Memory-bound part: global average pools (~924 MB streamed once -> ~40us at
// 23.3 TB/s).  These loads are marked non-temporal (TH=NT) so the one-shot
// 411MB/51MB feature maps don't thrash the 192MB L2 that the GEMM-chain
// weights/intermediates want to live in.
// Compute part: small fp32 GEMM chain done with V_WMMA_F32_16X16X4_F32.
// ---------------------------------------------------------------------------

typedef __attribute__((ext_vector_type(2))) float v2f;
typedef __attribute__((ext_vector_type(4))) float v4f;
typedef __attribute__((ext_vector_type(8))) float v8f;

#define B_   512
#define CI_  256
#define CO_  512
#define D_   64

__device__ __forceinline__ float wave_reduce_sum(float v) {
  #pragma unroll
  for (int off = 16; off > 0; off >>= 1)
    v += __shfl_down(v, off, 32);
  return v;
}

// ---------------- pooling: one wave per row --------------------------------
__global__ __launch_bounds__(256)
void pool_mean_vec4(const float* __restrict__ in, float* __restrict__ out,
                    int rows, int nvec4, float inv) {
  int wid  = threadIdx.x >> 5;
  int lane = threadIdx.x & 31;
  int row  = blockIdx.x * (blockDim.x >> 5) + wid;
  if (row >= rows) return;
  const v4f* p = (const v4f*)in + (size_t)row * nvec4;
  float s = 0.f;
  for (int v = lane; v < nvec4; v += 32) {
    v4f x = __builtin_nontemporal_load(&p[v]);      // streaming: TH=NT, b128
    s += (x.x + x.y) + (x.z + x.w);
  }
  s = wave_reduce_sum(s);
  if (lane == 0) out[row] = s * inv;
}

__global__ __launch_bounds__(256)
void pool_mean_scalar(const float* __restrict__ in, float* __restrict__ out,
                      int rows, int L, float inv) {
  int wid  = threadIdx.x >> 5;
  int lane = threadIdx.x & 31;
  int row  = blockIdx.x * (blockDim.x >> 5) + wid;
  if (row >= rows) return;
  const float* p = in + (size_t)row * L;
  float s = 0.f;
  for (int j = lane; j < L; j += 32)
    s += __builtin_nontemporal_load(&p[j]);         // streaming: TH=NT
  s = wave_reduce_sum(s);
  if (lane == 0) out[row] = s * inv;
}

// ---------------- fp32 WMMA GEMM: C = act(A[M,K] @ W[N,K]^T) ---------------
// One 16x16 C tile per wave.  K stepped by 4 (V_WMMA_F32_16X16X4_F32).
// A frag (16x4, 2 VGPR): lanes 0-15 -> K={k,k+1}, lanes 16-31 -> K={k+2,k+3},
// lane%16 = M row.  B frag (4x16) mirrored with lane%16 = N col; since
// B = W^T, both fragments are contiguous float2 loads along K.
// Grid must give every wave a tile (no predication: EXEC all-1s for WMMA).
__global__ __launch_bounds__(256)
void gemm_wmma_f32(const float* __restrict__ A, const float* __restrict__ W,
                   float* __restrict__ C, int M, int N, int K, int act) {
  int wid  = threadIdx.x >> 5;
  int lane = threadIdx.x & 31;
  int tile = blockIdx.x * (blockDim.x >> 5) + wid;
  int ntN  = N >> 4;
  int tm   = tile / ntN;
  int tn   = tile - tm * ntN;
  int m0   = tm << 4;
  int n0   = tn << 4;
  int half = lane >> 4;      // selects K pair {0,1} vs {2,3}
  int lm   = lane & 15;

  const float* arow = A + (size_t)(m0 + lm) * K + 2 * half;
  const float* wrow = W + (size_t)(n0 + lm) * K + 2 * half;

  v8f c = {0.f, 0.f, 0.f, 0.f, 0.f, 0.f, 0.f, 0.f};
  for (int k = 0; k < K; k += 4) {
    v2f a = *(const v2f*)(arow + k);
    v2f b = *(const v2f*)(wrow + k);
    c = __builtin_amdgcn_wmma_f32_16x16x4_f32(
        /*neg_a=*/false, a, /*neg_b=*/false, b,
        /*c_mod=*/(short)0, c, /*reuse_a=*/false, /*reuse_b=*/false);
  }

  // C/D layout: VGPR v -> row m0+v (lanes 0-15) / m0+v+8 (lanes 16-31)
  float* crow = C + (size_t)(m0 + half * 8) * N + n0 + lm;
  #pragma unroll
  for (int v = 0; v < 8; ++v) {
    float x = c[v];
    if (act) x = (x > 0.f) ? x : 0.01f * x;
    crow[(size_t)v * N] = x;
  }
}

// ---------------- small tail kernels ---------------------------------------
// out[i,0:64]=x[i]; out[i,64:128]=x[i]-center   (grid=B, block=64)
__global__ void concat_center_k(const float* __restrict__ x,
                                const float* __restrict__ center,
                                float* __restrict__ out) {
  int i = blockIdx.x, d = threadIdx.x;
  float v = x[i * D_ + d];
  out[i * 2 * D_ + d]      = v;
  out[i * 2 * D_ + D_ + d] = v - center[d];
}

// out[i,0:64]=x[i]; out[i,64:128]=x[i]-protos[cat[i]]
__global__ void concat_sel_k(const float* __restrict__ x,
                             const float* __restrict__ protos,
                             const int* __restrict__ cat,
                             float* __restrict__ out) {
  int i = blockIdx.x, d = threadIdx.x;
  float v = x[i * D_ + d];
  out[i * 2 * D_ + d]      = v;
  out[i * 2 * D_ + D_ + d] = v - protos[cat[i] * D_ + d];
}

// sim[i,p] = -||tex[i]-protos[p]||^2 ; cat[i] = argmax_p sim  (first max)
__global__ void sim_argmax_k(const float* __restrict__ tex,
                             const float* __restrict__ protos,
                             float* __restrict__ sim, int* __restrict__ cat) {
  int i = blockIdx.x * blockDim.x + threadIdx.x;
  if (i >= B_) return;
  const float* t = tex + i * D_;
  float best = -3.4e38f; int bi = 0;
  #pragma unroll
  for (int p = 0; p < 4; ++p) {
    float s = 0.f;
    for (int d = 0; d < D_; ++d) {
      float df = t[d] - protos[p * D_ + d];
      s += df * df;
    }
    float sv = -s;
    sim[i * 4 + p] = sv;
    if (sv > best) { best = sv; bi = p; }
  }
  cat[i] = bi;
}

// per-row SVDD distances to center
__global__ void svdd_rows_k(const float* __restrict__ cls,
                            const float* __restrict__ oc,
                            const float* __restrict__ center,
                            float* __restrict__ svdd_c,
                            float* __restrict__ svdd_o) {
  int i = blockIdx.x * blockDim.x + threadIdx.x;
  if (i >= B_) return;
  float sc = 0.f, so = 0.f;
  for (int d = 0; d < D_; ++d) {
    float cc = center[d];
    float a = cls[i * D_ + d] - cc;  sc += a * a;
    float b = oc[i * D_ + d]  - cc;  so += b * b;
  }
  svdd_c[i] = sc;
  svdd_o[i] = so;
}

__device__ float block_reduce_256(float v, float* red) {
  int t = threadIdx.x;
  red[t] = v;
  __syncthreads();
  #pragma unroll
  for (int s = 128; s > 0; s >>= 1) {
    if (t < s) red[t] += red[t + s];
    __syncthreads();
  }
  float r = red[0];
  __syncthreads();
  return r;
}

// distill loss + combined SVDD means -> out[4]
__global__ __launch_bounds__(256)
void final_reduce_k(const float* __restrict__ sim0, const float* __restrict__ sim1,
                    const float* __restrict__ so0, const float* __restrict__ sc0,
                    const float* __restrict__ so1, const float* __restrict__ sc1,
                    const int* __restrict__ epoch_ptr, float* __restrict__ out) {
  __shared__ float red[256];
  int ep = *epoch_ptr;
  // _SCHED: linspace(0.7,0.4,7) then 0.4  ->  0.7 - 0.05*ep for ep<7
  float temp = (ep < 7) ? (0.7f - 0.05f * (float)ep) : 0.4f;
  float itemp = 1.0f / temp;

  float l01 = 0.f, l10 = 0.f, ao = 0.f, ac = 0.f, al = 0.f;
  for (int i = threadIdx.x; i < B_; i += 256) {
    float s0[4], s1[4];
    #pragma unroll
    for (int j = 0; j < 4; ++j) { s0[j] = sim0[i * 4 + j]; s1[j] = sim1[i * 4 + j]; }
    float m0 = s0[0], m1 = s1[0];
    #pragma unroll
    for (int j = 1; j < 4; ++j) { m0 = fmaxf(m0, s0[j]); m1 = fmaxf(m1, s1[j]); }
    float z0t = 0.f, z1t = 0.f, z0s = 0.f, z1s = 0.f;
    #pragma unroll
    for (int j = 0; j < 4; ++j) {
      z0t += __expf((s0[j] - m0) * itemp);
      z1t += __expf((s1[j] - m1) * itemp);
      z0s += __expf(s0[j] - m0);
      z1s += __expf(s1[j] - m1);
    }
    float lse0 = m0 + __logf(z0s);
    float lse1 = m1 + __logf(z1s);
    #pragma unroll
    for (int j = 0; j < 4; ++j) {
      float q0 = __expf((s0[j] - m0) * itemp) / z0t;   // teacher from view 0
      float q1 = __expf((s1[j] - m1) * itemp) / z1t;   // teacher from view 1
      l01 += q0 * (lse1 - s1[j]);                      // -q0 * log_softmax(s1)
      l10 += q1 * (lse0 - s0[j]);
    }
    ao += so0[i] + so1[i];
    ac += sc0[i] + sc1[i];
    al += fabsf(so0[i] - sc0[i]) + fabsf(so1[i] - sc1[i]);
  }

  float L01 = block_reduce_256(l01, red);
  float L10 = block_reduce_256(l10, red);
  float AO  = block_reduce_256(ao,  red);
  float AC  = block_reduce_256(ac,  red);
  float AL  = block_reduce_256(al,  red);

  if (threadIdx.x == 0) {
    const float invB = 1.0f / (float)B_;
    out[0] = 0.5f * (L01 + L10) * invB;   // distill
    out[1] = AO * invB;                   // origin_svdd means (both branches)
    out[2] = AC * invB;                   // class_svdd means
    out[3] = AL * invB;                   // align means
  }
}

// ---------------------------------------------------------------------------
extern "C" void kernel_launch(void* const* d_in, const int* in_sizes, int n_in,
                              void* d_out, int out_size, void* d_ws, size_t ws_size,
                              hipStream_t stream) {
  (void)in_sizes; (void)n_in; (void)out_size; (void)ws_size;

  const float* inter[2]  = { (const float*)d_in[0], (const float*)d_in[2] };
  const float* origin[2] = { (const float*)d_in[1], (const float*)d_in[3] };
  const float* conv_w = (const float*)d_in[4];  // [512,256]
  const float* ofc_w1 = (const float*)d_in[5];  // [1024,512]
  const float* ofc_w2 = (const float*)d_in[6];  // [512,1024]
  const float* ofc_w3 = (const float*)d_in[7];  // [64,512]
  const float* sfc_w1 = (const float*)d_in[8];
  const float* sfc_w2 = (const float*)d_in[9];
  const float* sfc_w3 = (const float*)d_in[10];
  const float* tfc_w1 = (const float*)d_in[11]; // [64,128]
  const float* tfc_w2 = (const float*)d_in[12]; // [64,64]
  const float* cfc_w1 = (const float*)d_in[13];
  const float* cfc_w2 = (const float*)d_in[14];
  const float* oc_w1  = (const float*)d_in[15];
  const float* oc_w2  = (const float*)d_in[16];
  const float* center = (const float*)d_in[17];
  const float* protos = (const float*)d_in[18];
  const int*   epoch  = (const int*)d_in[19];
  float* out = (float*)d_out;

  // workspace layout (floats)
  float* ws = (float*)d_ws;
  size_t off = 0;
  float* pooled_ci = ws + off; off += (size_t)B_ * CI_;        // 512*256
  float* pooled_co = ws + off; off += (size_t)B_ * CO_;        // 512*512
  float* bufA      = ws + off; off += (size_t)B_ * 1024;       // [512,1024]
  float* bufB      = ws + off; off += (size_t)B_ * 512;        // [512,512]
  float* sh64      = ws + off; off += (size_t)B_ * D_;
  float* or64      = ws + off; off += (size_t)B_ * D_;
  float* cat128    = ws + off; off += (size_t)B_ * 2 * D_;
  float* t1        = ws + off; off += (size_t)B_ * D_;
  float* tex       = ws + off; off += (size_t)B_ * D_;
  float* cls64     = ws + off; off += (size_t)B_ * D_;
  float* oc64      = ws + off; off += (size_t)B_ * D_;
  float* sim_r[2];
  sim_r[0] = ws + off; off += (size_t)B_ * 4;
  sim_r[1] = ws + off; off += (size_t)B_ * 4;
  float* svdd_o[2]; float* svdd_c[2];
  svdd_o[0] = ws + off; off += B_;
  svdd_o[1] = ws + off; off += B_;
  svdd_c[0] = ws + off; off += B_;
  svdd_c[1] = ws + off; off += B_;
  int* cat_idx = (int*)(ws + off); off += B_;

  const dim3 blk256(256), blk64(64);
  auto gemm = [&](const float* A, const float* W, float* C,
                  int M, int N, int K, int act) {
    int tiles = (M >> 4) * (N >> 4);      // all cases divisible by 8 waves
    gemm_wmma_f32<<<dim3(tiles / 8), blk256, 0, stream>>>(A, W, C, M, N, K, act);
  };

  for (int r = 0; r < 2; ++r) {
    // ---- global average pools (the bandwidth-dominant work) ----
    pool_mean_vec4<<<dim3((B_ * CI_) / 8), blk256, 0, stream>>>(
        inter[r], pooled_ci, B_ * CI_, 784 / 4, 1.0f / 784.0f);
    pool_mean_scalar<<<dim3((B_ * CO_) / 8), blk256, 0, stream>>>(
        origin[r], pooled_co, B_ * CO_, 49, 1.0f / 49.0f);

    // ---- shallow branch: 1x1-conv (as GEMM after pool) + sfc MLP ----
    gemm(pooled_ci, conv_w, bufB, B_, CO_, CI_, 0);        // [512,512]
    gemm(bufB, sfc_w1, bufA, B_, 1024, 512, 1);
    gemm(bufA, sfc_w2, bufB, B_, 512, 1024, 1);
    gemm(bufB, sfc_w3, sh64, B_, 64, 512, 1);

    // ---- origin branch: ofc MLP ----
    gemm(pooled_co, ofc_w1, bufA, B_, 1024, 512, 1);
    gemm(bufA, ofc_w2, bufB, B_, 512, 1024, 1);
    gemm(bufB, ofc_w3, or64, B_, 64, 512, 1);

    // ---- texture -> sim / argmax ----
    concat_center_k<<<dim3(B_), blk64, 0, stream>>>(sh64, center, cat128);
    gemm(cat128, tfc_w1, t1, B_, 64, 128, 1);
    gemm(t1, tfc_w2, tex, B_, 64, 64, 1);
    sim_argmax_k<<<dim3(2), blk256, 0, stream>>>(tex, protos, sim_r[r], cat_idx);

    // ---- class feature ----
    concat_sel_k<<<dim3(B_), blk64, 0, stream>>>(or64, protos, cat_idx, cat128);
    gemm(cat128, cfc_w1, t1, B_, 64, 128, 1);
    gemm(t1, cfc_w2, cls64, B_, 64, 64, 1);

    // ---- origin classifier (oc) ----
    gemm(or64, oc_w1, t1, B_, 64, 64, 1);
    gemm(t1, oc_w2, oc64, B_, 64, 64, 1);

    // ---- per-row SVDD ----
    svdd_rows_k<<<dim3(2), blk256, 0, stream>>>(cls64, oc64, center,
                                                svdd_c[r], svdd_o[r]);
  }

  final_reduce_k<<<dim3(1), blk256, 0, stream>>>(
      sim_r[0], sim_r[1], svdd_o[0], svdd_c[0], svdd_o[1], svdd_c[1],
      epoch, out);
}